// Net_11390253269732
// MI455X (gfx1250) — compile-verified
//
#include <hip/hip_runtime.h>
#include <hip/hip_bf16.h>

#define HID 32

typedef float v8f __attribute__((ext_vector_type(8)));
typedef float v2f __attribute__((ext_vector_type(2)));

// ---------------------------------------------------------------------------
// Degree: deg starts at 1.0 (self loop), then atomic count of edge dst's.
// ---------------------------------------------------------------------------
__global__ __launch_bounds__(256) void deg_init_kernel(float* __restrict__ deg, int n) {
    int i = blockIdx.x * blockDim.x + threadIdx.x;
    if (i < n) deg[i] = 1.0f;
}

__global__ __launch_bounds__(256) void deg_accum_kernel(const int* __restrict__ dst,
                                                        float* __restrict__ deg, int nE) {
    int e = blockIdx.x * blockDim.x + threadIdx.x;
    if (e < nE) atomicAdd(&deg[dst[e]], 1.0f);
}

__global__ __launch_bounds__(256) void dinv_kernel(float* __restrict__ deg, int n) {
    int i = blockIdx.x * blockDim.x + threadIdx.x;
    if (i < n) {
        float d = deg[i];
        deg[i] = (d > 0.0f) ? rsqrtf(d) : 0.0f;   // in-place: deg -> dinv
    }
}

// ---------------------------------------------------------------------------
// fc1: h[n][j] = relu(x[n][0]*W[0][j] + x[n][1]*W[1][j] + b[j])   (K = 2)
// ---------------------------------------------------------------------------
__global__ __launch_bounds__(256) void fc1_kernel(const float* __restrict__ x,
                                                  const float* __restrict__ W,
                                                  const float* __restrict__ b,
                                                  float* __restrict__ h, int n) {
    int idx = blockIdx.x * blockDim.x + threadIdx.x;
    if (idx >= n * HID) return;
    int node = idx >> 5;
    int j    = idx & 31;
    float v = fmaf(x[node * 2 + 0], W[j], fmaf(x[node * 2 + 1], W[HID + j], b[j]));
    h[idx] = fmaxf(v, 0.0f);
}

// ---------------------------------------------------------------------------
// WMMA GEMM: out[N x 32] = A[N x 32] @ W[32 x 32]  (+ optional bias/ReLU)
// One wave per 16-row tile; 2 column tiles of 16; K stepped by 4 with
// v_wmma_f32_16x16x4_f32  (16 WMMA per wave).
//
// Fragment layouts per CDNA5 ISA 7.12.2 (wave32):
//   A 16x4 f32 : lane m = L%16, kh = L/16; VGPR0 = A[m][2kh], VGPR1 = A[m][2kh+1]
//   B  4x16 f32: lane n = L%16, kh = L/16; VGPR0 = B[2kh][n], VGPR1 = B[2kh+1][n]
//   C/D 16x16  : lane n = L%16; VGPR r holds row m = r + 8*(L/16)
// ---------------------------------------------------------------------------
__global__ __launch_bounds__(256) void gemm32_wmma_kernel(const float* __restrict__ A,
                                                          const float* __restrict__ W,
                                                          const float* __restrict__ bias,
                                                          float* __restrict__ out,
                                                          int nrows, int apply_relu) {
    const int lane = threadIdx.x & 31;
    const int tile = ((blockIdx.x * blockDim.x + threadIdx.x) >> 5);  // 16-row tile id
    const int row0 = tile * 16;
    if (row0 >= nrows) return;                 // wave-uniform exit

    const int m  = lane & 15;                  // A row within tile / D column
    const int kh = lane >> 4;                  // K-half select (0/1)
    int arow = row0 + m;
    if (arow >= nrows) arow = nrows - 1;       // clamp loads; guarded on store

    // Preload all B fragments (weights): 8 K-steps x 2 N-tiles.
    v2f bfrag[8][2];
#pragma unroll
    for (int t = 0; t < 8; ++t) {
        const int kr = 4 * t + 2 * kh;
#pragma unroll
        for (int j = 0; j < 2; ++j) {
            bfrag[t][j].x = W[(kr + 0) * HID + 16 * j + m];
            bfrag[t][j].y = W[(kr + 1) * HID + 16 * j + m];
        }
    }

    v8f acc0 = {};
    v8f acc1 = {};
    const float* arowp = A + (size_t)arow * HID;
#pragma unroll
    for (int t = 0; t < 8; ++t) {
        const int k = 4 * t + 2 * kh;
        v2f a;
        a.x = arowp[k];
        a.y = arowp[k + 1];
        acc0 = __builtin_amdgcn_wmma_f32_16x16x4_f32(false, a, false, bfrag[t][0],
                                                     (short)0, acc0, false, false);
        acc1 = __builtin_amdgcn_wmma_f32_16x16x4_f32(false, a, false, bfrag[t][1],
                                                     (short)0, acc1, false, false);
    }

    // Epilogue + store (per-lane guards are fine here: no more WMMA).
#pragma unroll
    for (int r = 0; r < 8; ++r) {
        const int mr = row0 + r + 8 * kh;
        if (mr < nrows) {
            float v0 = acc0[r];
            float v1 = acc1[r];
            if (bias) { v0 += bias[m]; v1 += bias[16 + m]; }
            if (apply_relu) { v0 = fmaxf(v0, 0.0f); v1 = fmaxf(v1, 0.0f); }
            out[(size_t)mr * HID + m]      = v0;
            out[(size_t)mr * HID + 16 + m] = v1;
        }
    }
}

// ---------------------------------------------------------------------------
// Self-loop term initializes the accumulator: acc[v][:] = hW[v][:] * dinv[v]^2
// (also serves as the zero-init of acc — no memset, no 100K extra atomics)
// ---------------------------------------------------------------------------
__global__ __launch_bounds__(256) void selfloop_init_kernel(const float* __restrict__ hW,
                                                            const float* __restrict__ dinv,
                                                            float* __restrict__ acc, int n) {
    int idx = blockIdx.x * blockDim.x + threadIdx.x;
    if (idx >= n * HID) return;
    float di = dinv[idx >> 5];
    acc[idx] = hW[idx] * di * di;
}

// ---------------------------------------------------------------------------
// Edge scatter: one wave32 per edge, lane = feature.
//   gather 128B row of hW[src], scale by dinv[src]*dinv[dst],
//   global_atomic_add_f32 into acc[dst] (L2-resident).
// ---------------------------------------------------------------------------
__global__ __launch_bounds__(256) void edge_scatter_kernel(const float* __restrict__ hW,
                                                           const float* __restrict__ dinv,
                                                           const int* __restrict__ src,
                                                           const int* __restrict__ dst,
                                                           float* __restrict__ acc, int nE) {
    int gid  = blockIdx.x * blockDim.x + threadIdx.x;
    int e    = gid >> 5;
    int lane = gid & 31;
    if (e >= nE) return;
    int s = src[e];                 // wave-uniform -> scalar loads
    int d = dst[e];
    float norm = dinv[s] * dinv[d];
    float v = hW[(size_t)s * HID + lane] * norm;
    atomicAdd(&acc[(size_t)d * HID + lane], v);
}

// ---------------------------------------------------------------------------
// Bias + ReLU epilogue for conv layers: h = relu(acc + b)
// ---------------------------------------------------------------------------
__global__ __launch_bounds__(256) void bias_relu_kernel(const float* __restrict__ acc,
                                                        const float* __restrict__ b,
                                                        float* __restrict__ h, int n) {
    int idx = blockIdx.x * blockDim.x + threadIdx.x;
    if (idx >= n * HID) return;
    h[idx] = fmaxf(acc[idx] + b[idx & 31], 0.0f);
}

// ---------------------------------------------------------------------------
// fc3: out[n] = dot(h[n][:], w) + b  — one wave per node, shfl_xor reduce.
// ---------------------------------------------------------------------------
__global__ __launch_bounds__(256) void fc3_kernel(const float* __restrict__ h,
                                                  const float* __restrict__ w,
                                                  const float* __restrict__ b,
                                                  float* __restrict__ out, int n) {
    int gid  = blockIdx.x * blockDim.x + threadIdx.x;
    int node = gid >> 5;
    int lane = gid & 31;
    if (node >= n) return;
    float v = h[(size_t)node * HID + lane] * w[lane];
#pragma unroll
    for (int off = 16; off > 0; off >>= 1) v += __shfl_xor(v, off, 32);
    if (lane == 0) out[node] = v + b[0];
}

// ---------------------------------------------------------------------------
// Launch
// ---------------------------------------------------------------------------
static inline unsigned cdiv(long long a, long long b) { return (unsigned)((a + b - 1) / b); }

extern "C" void kernel_launch(void* const* d_in, const int* in_sizes, int n_in,
                              void* d_out, int out_size, void* d_ws, size_t ws_size,
                              hipStream_t stream) {
    const float* x       = (const float*)d_in[0];
    const int*   edges   = (const int*)  d_in[1];
    const float* fc1_W   = (const float*)d_in[2];
    const float* fc1_b   = (const float*)d_in[3];
    const float* c1_W    = (const float*)d_in[4];
    const float* c1_b    = (const float*)d_in[5];
    const float* c2_W    = (const float*)d_in[6];
    const float* c2_b    = (const float*)d_in[7];
    const float* c3_W    = (const float*)d_in[8];
    const float* c3_b    = (const float*)d_in[9];
    const float* fc2_W   = (const float*)d_in[10];
    const float* fc2_b   = (const float*)d_in[11];
    const float* fc3_W   = (const float*)d_in[12];
    const float* fc3_b   = (const float*)d_in[13];
    float*       out     = (float*)d_out;

    const int NN = in_sizes[0] / 2;     // nodes (x is [N,2])
    const int NE = in_sizes[1] / 2;     // edges (edge_index is [2,E])
    const int* src = edges;
    const int* dst = edges + NE;

    // Workspace layout (all L2-resident: ~39 MB total)
    float* bufA = (float*)d_ws;               // h   [NN,32]
    float* bufB = bufA + (size_t)NN * HID;    // hW  [NN,32]
    float* bufC = bufB + (size_t)NN * HID;    // acc [NN,32]
    float* dinv = bufC + (size_t)NN * HID;    // deg -> dinv [NN]

    const unsigned T = 256;
    const unsigned gNode  = cdiv(NN, T);
    const unsigned gEdge  = cdiv(NE, T);
    const unsigned gFeat  = cdiv((long long)NN * HID, T);
    const unsigned gEdgeW = cdiv((long long)NE * HID, T);   // wave per edge
    const unsigned gGemm  = cdiv((long long)cdiv(NN, 16) * 32, T); // wave per 16-row tile
    const unsigned gDot   = cdiv((long long)NN * HID, T);   // wave per node

    // 1) degrees -> dinv
    deg_init_kernel<<<gNode, T, 0, stream>>>(dinv, NN);
    deg_accum_kernel<<<gEdge, T, 0, stream>>>(dst, dinv, NE);
    dinv_kernel<<<gNode, T, 0, stream>>>(dinv, NN);

    // 2) fc1
    fc1_kernel<<<gFeat, T, 0, stream>>>(x, fc1_W, fc1_b, bufA, NN);

    // 3) three GCN conv layers
    const float* convW[3] = {c1_W, c2_W, c3_W};
    const float* convB[3] = {c1_b, c2_b, c3_b};
    for (int l = 0; l < 3; ++l) {
        gemm32_wmma_kernel<<<gGemm, T, 0, stream>>>(bufA, convW[l], nullptr, bufB, NN, 0);
        selfloop_init_kernel<<<gFeat, T, 0, stream>>>(bufB, dinv, bufC, NN);
        edge_scatter_kernel<<<gEdgeW, T, 0, stream>>>(bufB, dinv, src, dst, bufC, NE);
        bias_relu_kernel<<<gFeat, T, 0, stream>>>(bufC, convB[l], bufA, NN);
    }

    // 4) fc2 (fused bias+relu epilogue) and fc3
    gemm32_wmma_kernel<<<gGemm, T, 0, stream>>>(bufA, fc2_W, fc2_b, bufB, NN, 1);
    fc3_kernel<<<gDot, T, 0, stream>>>(bufB, fc3_W, fc3_b, out, NN);
}